// OrdinalRegressionLoss_3126736191980
// MI455X (gfx1250) — compile-verified
//
#include <hip/hip_runtime.h>

typedef __attribute__((ext_vector_type(2))) float v2f;
typedef __attribute__((ext_vector_type(8))) float v8f;

#define BLOCK 256
#define GRID  1024

// Exact 32-lane sum using the CDNA5 matrix pipe:
// A(16x4): A[m][0] = p_m (lanes 0-15, VGPR0), A[m][2] = p_{m+16} (lanes 16-31, VGPR0),
//          A[m][1] = A[m][3] = 0 (VGPR1).
// B(4x16) = ones, C = 0  =>  D[m][n] = p_m + p_{m+16} (independent of n).
// D layout: lane L holds column N=L%16; VGPR v holds M=v (lanes 0-15) or M=v+8 (lanes 16-31).
// Sum of a lane's 8 D regs = half-sum over M; shfl_xor(16) merges halves -> total in all lanes.
__device__ __forceinline__ float wave_sum_wmma(float p) {
    v2f a; a[0] = p;    a[1] = 0.0f;   // A-matrix fragment (2 VGPRs)
    v2f b; b[0] = 1.0f; b[1] = 1.0f;   // B-matrix = ones  (2 VGPRs)
    v8f c = {};                        // C = 0
    v8f d = __builtin_amdgcn_wmma_f32_16x16x4_f32(
        /*neg_a=*/false, a, /*neg_b=*/false, b,
        /*c_mod=*/(short)0, c, /*reuse_a=*/false, /*reuse_b=*/false);
    float s = ((d[0] + d[1]) + (d[2] + d[3])) + ((d[4] + d[5]) + (d[6] + d[7]));
    s += __shfl_xor(s, 16, 32);
    return s;
}

// Stage 1: streaming BCE-with-logits over ordinal targets, one partial per block.
// Per row i: loss = sum_j [ max(x_j,0) + log1p(exp(-|x_j|)) ] - sum_{j<t} x_j
__global__ void ord_loss_partial(const float4* __restrict__ logits,
                                 const int*    __restrict__ targets,
                                 float*        __restrict__ partial,
                                 int nrows) {
    const int tid    = blockIdx.x * blockDim.x + threadIdx.x;
    const int stride = gridDim.x * blockDim.x;

    float acc = 0.0f;
    for (int i = tid; i < nrows; i += stride) {
        const float4 x = logits[i];       // global_load_b128
        const int    t = targets[i];      // global_load_b32

        float s;
        s  = fmaxf(x.x, 0.0f) + __logf(1.0f + __expf(-fabsf(x.x)));
        s += fmaxf(x.y, 0.0f) + __logf(1.0f + __expf(-fabsf(x.y)));
        s += fmaxf(x.z, 0.0f) + __logf(1.0f + __expf(-fabsf(x.z)));
        s += fmaxf(x.w, 0.0f) + __logf(1.0f + __expf(-fabsf(x.w)));
        // -x*y with y_j = (j < t): branchless selects keep EXEC full.
        s -= (t > 0) ? x.x : 0.0f;
        s -= (t > 1) ? x.y : 0.0f;
        s -= (t > 2) ? x.z : 0.0f;
        s -= (t > 3) ? x.w : 0.0f;
        acc += s;
    }

    // Wave-level exact sum via V_WMMA_F32_16X16X4_F32 (EXEC reconverged = all 1s here).
    const float wsum = wave_sum_wmma(acc);

    __shared__ float lds[BLOCK / 32];
    const int lane = threadIdx.x & 31;
    const int wave = threadIdx.x >> 5;
    if (lane == 0) lds[wave] = wsum;
    __syncthreads();

    if (threadIdx.x == 0) {
        float b = 0.0f;
        #pragma unroll
        for (int w = 0; w < BLOCK / 32; ++w) b += lds[w];
        partial[blockIdx.x] = b;   // every block overwrites its slot: deterministic
    }
}

// Stage 2: single full wave folds the block partials; WMMA again for the 32->1 step.
__global__ void ord_loss_final(const float* __restrict__ partial, int n,
                               float* __restrict__ out, float inv_count) {
    float acc = 0.0f;
    for (int i = threadIdx.x; i < n; i += 32) acc += partial[i];
    const float tot = wave_sum_wmma(acc);
    if (threadIdx.x == 0) out[0] = tot * inv_count;   // mean; plain store (idempotent)
}

extern "C" void kernel_launch(void* const* d_in, const int* in_sizes, int n_in,
                              void* d_out, int out_size, void* d_ws, size_t ws_size,
                              hipStream_t stream) {
    const float4* logits  = (const float4*)d_in[0];  // (B, 4) float32, row = float4
    const int*    targets = (const int*)d_in[1];     // (B,) integer
    const int     nrows   = in_sizes[1];             // B

    float* partial = (float*)d_ws;                   // GRID floats of scratch

    ord_loss_partial<<<GRID, BLOCK, 0, stream>>>(logits, targets, partial, nrows);

    const float inv_count = 1.0f / (4.0f * (float)nrows);
    ord_loss_final<<<1, 32, 0, stream>>>(partial, GRID, (float*)d_out, inv_count);
}